// Voxelizer_69020124446919
// MI455X (gfx1250) — compile-verified
//
#include <hip/hip_runtime.h>
#include <hip/hip_bf16.h>

typedef __attribute__((ext_vector_type(2))) float v2f;
typedef __attribute__((ext_vector_type(8))) float v8f;

#define DD 128
#define HH 128
#define WW 128
#define CUTOFF 3.0f
#define HALF 64.0f
#define INV_HALF (1.0f / 64.0f)

__global__ void vox_zero_kernel(float* __restrict__ out, int n) {
    int i = blockIdx.x * blockDim.x + threadIdx.x;
    int stride = gridDim.x * blockDim.x;
    for (; i < n; i += stride) out[i] = 0.0f;
}

__global__ __launch_bounds__(256) void vox_splat_kernel(
    const float* __restrict__ pos,   // N x 3
    const float* __restrict__ scl,   // N x 3
    const float* __restrict__ rot,   // N x 4  (w,x,y,z)
    const float* __restrict__ den,   // N
    float* __restrict__ vol,         // D*H*W
    int n) {
    const int lane = threadIdx.x & 31;
    const int wave = threadIdx.x >> 5;
    const int g = blockIdx.x * 8 + wave;
    if (g >= n) return;  // lane-uniform value -> surviving waves keep EXEC all-1s

    // ---- per-Gaussian preamble (lane-uniform values) ----
    float qw = rot[g * 4 + 0], qx = rot[g * 4 + 1];
    float qy = rot[g * 4 + 2], qz = rot[g * 4 + 3];
    float qn = rsqrtf(qw * qw + qx * qx + qy * qy + qz * qz);
    qw *= qn; qx *= qn; qy *= qn; qz *= qn;

    const float r00 = 1.0f - 2.0f * (qy * qy + qz * qz);
    const float r01 = 2.0f * (qx * qy - qw * qz);
    const float r02 = 2.0f * (qx * qz + qw * qy);
    const float r10 = 2.0f * (qx * qy + qw * qz);
    const float r11 = 1.0f - 2.0f * (qx * qx + qz * qz);
    const float r12 = 2.0f * (qy * qz - qw * qx);
    const float r20 = 2.0f * (qx * qz - qw * qy);
    const float r21 = 2.0f * (qy * qz + qw * qx);
    const float r22 = 1.0f - 2.0f * (qx * qx + qy * qy);

    const float s0 = scl[g * 3 + 0], s1 = scl[g * 3 + 1], s2 = scl[g * 3 + 2];
    const float i0 = 1.0f / (s0 * s0 + 1e-8f);
    const float i1 = 1.0f / (s1 * s1 + 1e-8f);
    const float i2 = 1.0f / (s2 * s2 + 1e-8f);

    // Sigma^{-1} = R diag(i) R^T (symmetric)
    const float cA = r00 * r00 * i0 + r01 * r01 * i1 + r02 * r02 * i2;  // (d,d)
    const float cB = r10 * r10 * i0 + r11 * r11 * i1 + r12 * r12 * i2;  // (h,h)
    const float cC = r20 * r20 * i0 + r21 * r21 * i1 + r22 * r22 * i2;  // (w,w)
    const float cE = r00 * r10 * i0 + r01 * r11 * i1 + r02 * r12 * i2;  // (d,h)
    const float cF = r00 * r20 * i0 + r01 * r21 * i1 + r02 * r22 * i2;  // (d,w)
    const float cG = r10 * r20 * i0 + r11 * r21 * i1 + r12 * r22 * i2;  // (h,w)

    const float p0 = pos[g * 3 + 0], p1 = pos[g * 3 + 1], p2 = pos[g * 3 + 2];
    const float pv0 = (p0 + 1.0f) * HALF - 0.5f;
    const float pv1 = (p1 + 1.0f) * HALF - 0.5f;
    const float pv2 = (p2 + 1.0f) * HALF - 0.5f;
    const float radius = fmaxf(s0, fmaxf(s1, s2)) * HALF * CUTOFF;

    // Force wave-uniform control state into SGPRs: scalar loop (s_cbranch),
    // EXEC provably all-1s at the WMMA, SALU addressing (SADDR form).
    const int mn_d = __builtin_amdgcn_readfirstlane(
        min(max((int)floorf(pv0 - radius), 0), DD - 1));
    const int mn_h = __builtin_amdgcn_readfirstlane(
        min(max((int)floorf(pv1 - radius), 0), HH - 1));
    const int mn_w = __builtin_amdgcn_readfirstlane(
        min(max((int)floorf(pv2 - radius), 0), WW - 1));
    const int mx_d = __builtin_amdgcn_readfirstlane(
        min(max((int)ceilf(pv0 + radius), 0), DD - 1) + 1);
    const int mx_h = __builtin_amdgcn_readfirstlane(
        min(max((int)ceilf(pv1 + radius), 0), HH - 1) + 1);
    const int mx_w = __builtin_amdgcn_readfirstlane(
        min(max((int)ceilf(pv2 + radius), 0), WW - 1) + 1);
    const int nd = min(mx_d - mn_d, 16);
    const int nh = mx_h - mn_h;
    const int nw = mx_w - mn_w;
    const float dens = den[g];

    // ---- per-lane precompute ----
    const int t = lane & 15;          // h row (A) / w column (B,C)
    const bool lo = lane < 16;        // lanes 0-15: K=0,1 ; lanes 16-31: K=2,3
    const float dh = ((float)(mn_h + t) + 0.5f) * INV_HALF - 1.0f - p1;
    const float dw = ((float)(mn_w + t) + 0.5f) * INV_HALF - 1.0f - p2;

    // A fragment, constant over d-slices: A[h, 0..3] = [ 1, dh, b*dh^2, 0 ]
    v2f afrag;
    afrag[0] = lo ? 1.0f : cB * dh * dh;   // K=0 (lo) / K=2 (hi)
    afrag[1] = lo ? dh : 0.0f;             // K=1 (lo) / K=3 (hi)

    const float cw2 = cC * dw * dw;        // c*dw^2
    const float fw2 = 2.0f * cF * dw;      // 2f*dw
    const float gw2 = 2.0f * cG * dw;      // 2g*dw
    const float twoE = 2.0f * cE;

    const bool w_ok = (t < nw);
    const int h_add = lo ? 0 : 8;          // C/D layout: lanes>=16 hold M=8..15
    const int laneoff = h_add * WW + t;    // per-lane part of the address

    for (int s = 0; s < nd; ++s) {         // scalar loop: nd is in an SGPR
        const int id = mn_d + s;
        const float ddv = ((float)id + 0.5f) * INV_HALF - 1.0f - p0;

        // B per slice: B[0,w]=c*dw^2+2f*dd*dw+a*dd^2 ; B[1,w]=2g*dw+2e*dd ;
        //              B[2,w]=1 ; B[3,w]=0
        v2f bfrag;
        bfrag[0] = lo ? (cw2 + ddv * fw2 + cA * ddv * ddv) : 1.0f;
        bfrag[1] = lo ? (gw2 + twoE * ddv) : 0.0f;

        v8f acc = {};
        acc = __builtin_amdgcn_wmma_f32_16x16x4_f32(
            /*neg_a=*/false, afrag, /*neg_b=*/false, bfrag,
            /*c_mod=*/(short)0, acc, /*reuse_a=*/false, /*reuse_b=*/false);
        // acc[r] = mahal(h = r + h_add, w = t) for this slice

        // 1) all 8 transcendentals up-front so they pipeline on the trans unit
        float wgt[8];
#pragma unroll
        for (int r = 0; r < 8; ++r)
            wgt[r] = __expf(-0.5f * acc[r]) * dens;

        // 2) compact predicated atomic blocks (scalar base, imm offsets)
        float* slice = vol + ((id * HH + mn_h) * WW + mn_w);
#pragma unroll
        for (int r = 0; r < 8; ++r) {
            if (w_ok && (r + h_add) < nh) {
                atomicAdd(slice + laneoff + r * WW, wgt[r]);
            }
        }
    }
}

extern "C" void kernel_launch(void* const* d_in, const int* in_sizes, int n_in,
                              void* d_out, int out_size, void* d_ws, size_t ws_size,
                              hipStream_t stream) {
    const float* positions = (const float*)d_in[0];
    const float* scales    = (const float*)d_in[1];
    const float* rotations = (const float*)d_in[2];
    const float* density   = (const float*)d_in[3];
    float* vol = (float*)d_out;
    const int n = in_sizes[3];  // N gaussians (density is N elements)

    // zero the (poisoned) output volume
    int zb = (out_size + 255) / 256;
    if (zb > 2048) zb = 2048;
    vox_zero_kernel<<<zb, 256, 0, stream>>>(vol, out_size);

    // one wave (32 lanes) per gaussian, 8 waves per 256-thread block
    const int blocks = (n + 7) / 8;
    vox_splat_kernel<<<blocks, 256, 0, stream>>>(positions, scales, rotations,
                                                 density, vol, n);
}